// CustomCRF_39737037423422
// MI455X (gfx1250) — compile-verified
//
#include <hip/hip_runtime.h>
#include <hip/hip_bf16.h>

#define SEQ_LEN 1024
#define FOT_W   1.8f

// ---------------------------------------------------------------------------
// CDNA5 async global->LDS path (gfx1250). Guarded so the file compiles on the
// host pass and on toolchains lacking the builtins.
// ---------------------------------------------------------------------------
#if defined(__AMDGCN__) && \
    __has_builtin(__builtin_amdgcn_global_load_async_to_lds_b128) && \
    __has_builtin(__builtin_amdgcn_s_wait_asynccnt)
#define HAVE_ASYNC_LDS 1
#else
#define HAVE_ASYNC_LDS 0
#endif

#if HAVE_ASYNC_LDS
typedef int v4i_t __attribute__((vector_size(16)));
typedef __attribute__((address_space(1))) v4i_t* gv4i_p;   // global (AS1)
typedef __attribute__((address_space(3))) v4i_t* lv4i_p;   // LDS (AS3)
__device__ __forceinline__ void async_copy16(const void* g, void* l) {
    // per-lane 16B global -> LDS, tracked by ASYNCcnt
    __builtin_amdgcn_global_load_async_to_lds_b128(
        (gv4i_p)(void*)g, (lv4i_p)l, 0, 0);
}
#endif

// ---------------------------------------------------------------------------
// CDNA5 WMMA wave reduction: sum 32 lane values with one
// V_WMMA_F32_16X16X4_F32 (A = data in K0 with K1=0, B = all-ones so the
// undocumented B striping is irrelevant; D layout is documented).
// ---------------------------------------------------------------------------
#if defined(__AMDGCN__) && __has_builtin(__builtin_amdgcn_wmma_f32_16x16x4_f32)
#define HAVE_WMMA 1
#else
#define HAVE_WMMA 0
#endif

typedef float v8f_t __attribute__((ext_vector_type(8)));
typedef float v2f_t __attribute__((ext_vector_type(2)));

#if HAVE_WMMA
__device__ __forceinline__ float wave_sum_wmma(float v) {
    v2f_t a;            // A is 16x4 f32: VGPR0 = K0 (lanes 0-15) / K2 (lanes 16-31)
    a.x = v;            //   -> A[M][0] = val(lane M), A[M][2] = val(lane M+16)
    a.y = 0.f;          //   -> A[M][1] = A[M][3] = 0
    v2f_t bones;        // B = ones 4x16 (layout-invariant)
    bones.x = 1.f; bones.y = 1.f;
    v8f_t c = {};
    // D[M][N] = sum_K A[M][K] * 1 = val(lane M) + val(lane M+16), for all N
    v8f_t d = __builtin_amdgcn_wmma_f32_16x16x4_f32(
        /*neg_a=*/false, a, /*neg_b=*/false, bones,
        /*c_mod=*/(short)0, c, /*reuse_a=*/false, /*reuse_b=*/false);
    // VGPR r holds row M=r (lanes 0-15) and M=r+8 (lanes 16-31):
    // summing d[0..7] gives each half-wave the sum of its 16 source lanes.
    float s = d[0] + d[1] + d[2] + d[3] + d[4] + d[5] + d[6] + d[7];
    s += __shfl_xor(s, 16, 32);   // combine the two half-wave sums
    return s;                      // full wave sum in every lane
}
#endif

// Block (256 threads) sum; result valid on thread 0.
__device__ __forceinline__ float block_reduce256(float v) {
#if HAVE_WMMA
    __shared__ float wsum[8];
    int tid = threadIdx.x;
    float s = wave_sum_wmma(v);
    if ((tid & 31) == 0) wsum[tid >> 5] = s;
    __syncthreads();
    float t = 0.f;
    if (tid == 0)
        for (int i = 0; i < 8; ++i) t += wsum[i];
    return t;
#else
    __shared__ float red[256];
    int tid = threadIdx.x;
    red[tid] = v;
    __syncthreads();
    for (int off = 128; off > 0; off >>= 1) {
        if (tid < off) red[tid] += red[tid + off];
        __syncthreads();
    }
    return red[0];
#endif
}

// ---------------------------------------------------------------------------
// Kernel A: gold score + constraint loss, fully parallel over (b,s).
// Accumulates (-score_terms + 0.1*cl*m) per element.
// ---------------------------------------------------------------------------
__global__ void crf_score_constraint_kernel(
    const float* __restrict__ em,  const int* __restrict__ tg,
    const int* __restrict__ mk,    const int* __restrict__ pid,
    const int* __restrict__ the,   const int* __restrict__ intens,
    const int* __restrict__ lower, const float* __restrict__ pw,
    const float* __restrict__ T,   const float* __restrict__ startT,
    const float* __restrict__ endT, float* __restrict__ partial, int B)
{
    __shared__ float sPW[17], sT[9], sS[3], sE[3];
    int tid = threadIdx.x;
    if (tid < 17) sPW[tid] = pw[tid];
    if (tid < 9)  sT[tid]  = T[tid];
    if (tid < 3)  { sS[tid] = startT[tid]; sE[tid] = endT[tid]; }
    __syncthreads();

    const long long N = (long long)B * SEQ_LEN;
    const long long stride = (long long)gridDim.x * blockDim.x;
    float acc = 0.f;
    for (long long idx = (long long)blockIdx.x * blockDim.x + tid; idx < N; idx += stride) {
        __builtin_prefetch(tg + idx + stride, 0, 0);
        __builtin_prefetch(em + (idx + stride) * 3, 0, 0);
        int   s  = (int)(idx & (SEQ_LEN - 1));
        int   t  = tg[idx];
        float m  = (float)mk[idx];
        int   p  = pid[idx];
        float fw = (t == 0) ? 1.0f : FOT_W;
        float ew = em[idx * 3 + t] * sPW[p] * fw;   // weighted emission at gold tag
        float sc;
        if (s == 0) {
            sc = sS[t] + ew;                        // start term is unmasked
        } else {
            int tp = tg[idx - 1];
            sc = (sT[tp * 3 + t] + ew) * m;
            // constraint loss terms (positions 1..S-1)
            bool cur1 = (t == 1), cur2 = (t == 2);
            bool infot = cur1 | cur2;
            bool nextO = (s == SEQ_LEN - 1) ? true : (tg[idx + 1] == 0);
            float cl = 0.f;
            cl += 0.5f * (float)((tp == 1) & !cur2);
            cl += 1.0f * (float)(cur2 & !((tp == 1) | (tp == 2)));
            cl += 0.8f * (float)(((p == 2) | (p == 3)) & cur1);           // ADP|CCONJ
            cl += 0.8f * (float)((p == 4) & infot & (the[idx] == 0));     // DET & ~the
            cl += 1.0f * (float)((p == 5) & infot);                       // AUX
            cl += 0.8f * (float)((p == 6) & (intens[idx] != 0) & cur1);   // ADV & intens
            cl += 0.9f * (float)((p == 7) & infot);                       // PRON
            cl += 0.7f * (float)((p == 8) & cur1 & nextO);                // NUM
            cl += 0.5f * (float)(infot & (lower[idx] != 0) & !((p == 0) | (p == 1)));
            acc += 0.1f * cl * m;
        }
        // end transition: last valid step of a contiguous mask prefix
        bool isLast = (m != 0.f) && ((s == SEQ_LEN - 1) || (mk[idx + 1] == 0));
        if (isLast) sc += sE[t];
        acc -= sc;
    }
    float tot = block_reduce256(acc);
    if (tid == 0) partial[blockIdx.x] = tot;
}

// ---------------------------------------------------------------------------
// Kernel B: partition function. One batch row per lane (wave32), 3-state
// forward recursion in log space with exp(T) precomputed. Emissions staged
// through LDS with double-buffered async B128 copies.
// ---------------------------------------------------------------------------
__global__ void crf_partition_kernel(
    const float* __restrict__ em, const int* __restrict__ mk,
    const int* __restrict__ pid,  const float* __restrict__ pw,
    const float* __restrict__ T,  const float* __restrict__ startT,
    const float* __restrict__ endT, float* __restrict__ partial,
    int B, int partialOffset)
{
    int tid = threadIdx.x;
    int b   = blockIdx.x * blockDim.x + tid;
    bool active = (b < B);
    int  bb = active ? b : (B - 1);

    __shared__ float sPW[17], sExpT[9], sS[3], sE[3];
    if (tid < 17) sPW[tid]   = pw[tid];
    if (tid < 9)  sExpT[tid] = __expf(T[tid]);
    if (tid < 3)  { sS[tid] = startT[tid]; sE[tid] = endT[tid]; }
#if HAVE_ASYNC_LDS
    constexpr int TS = 8;                                   // timesteps per tile
    __shared__ __align__(16) float stage[2][256 * TS * 3];  // 2 x 24 KB
#endif
    __syncthreads();

    const float* gb = em  + (long long)bb * SEQ_LEN * 3;
    const int*   gm = mk  + (long long)bb * SEQ_LEN;
    const int*   gp = pid + (long long)bb * SEQ_LEN;

    float a0 = 0.f, a1 = 0.f, a2 = 0.f;

#define CRF_STEP(S_, E0_, E1_, E2_)                                          \
    do {                                                                     \
        float w_  = sPW[gp[S_]];                                             \
        float e0_ = (E0_) * w_;                                              \
        float e1_ = (E1_) * w_ * FOT_W;                                      \
        float e2_ = (E2_) * w_ * FOT_W;                                      \
        if ((S_) == 0) {                                                     \
            a0 = sS[0] + e0_; a1 = sS[1] + e1_; a2 = sS[2] + e2_;            \
        } else {                                                             \
            float mx_ = fmaxf(a0, fmaxf(a1, a2));                            \
            float p0_ = __expf(a0 - mx_);                                    \
            float p1_ = __expf(a1 - mx_);                                    \
            float p2_ = __expf(a2 - mx_);                                    \
            float q0_ = p0_*sExpT[0] + p1_*sExpT[3] + p2_*sExpT[6];          \
            float q1_ = p0_*sExpT[1] + p1_*sExpT[4] + p2_*sExpT[7];          \
            float q2_ = p0_*sExpT[2] + p1_*sExpT[5] + p2_*sExpT[8];          \
            float n0_ = e0_ + mx_ + __logf(q0_);                             \
            float n1_ = e1_ + mx_ + __logf(q1_);                             \
            float n2_ = e2_ + mx_ + __logf(q2_);                             \
            if (gm[S_]) { a0 = n0_; a1 = n1_; a2 = n2_; }                    \
        }                                                                    \
    } while (0)

#if HAVE_ASYNC_LDS
    float* st0 = &stage[0][tid * (TS * 3)];
    float* st1 = &stage[1][tid * (TS * 3)];
    // prologue: stage tile 0 (TS*3*4 = 96 B per lane = 6 x b128)
    for (int c = 0; c < 6; ++c)
        async_copy16(gb + c * 4, st0 + c * 4);
    int buf = 0;
    for (int s0 = 0; s0 < SEQ_LEN; s0 += TS) {
        float* cur = buf ? st1 : st0;
        float* nxt = buf ? st0 : st1;
        if (s0 + TS < SEQ_LEN) {
            __builtin_prefetch(gp + s0 + TS, 0, 0);
            __builtin_prefetch(gm + s0 + TS, 0, 0);
            for (int c = 0; c < 6; ++c)
                async_copy16(gb + (s0 + TS) * 3 + c * 4, nxt + c * 4);
            __builtin_amdgcn_s_wait_asynccnt(6);   // current tile complete (in-order)
        } else {
            __builtin_amdgcn_s_wait_asynccnt(0);   // drain for last tile
        }
        for (int i = 0; i < TS; ++i) {
            int s = s0 + i;
            CRF_STEP(s, cur[i * 3 + 0], cur[i * 3 + 1], cur[i * 3 + 2]);
        }
        buf ^= 1;
    }
#else
    for (int s = 0; s < SEQ_LEN; ++s) {
        __builtin_prefetch(gb + (s + 16) * 3, 0, 0);
        const float* ep = gb + s * 3;
        CRF_STEP(s, ep[0], ep[1], ep[2]);
    }
#endif
#undef CRF_STEP

    float z0 = a0 + sE[0], z1 = a1 + sE[1], z2 = a2 + sE[2];
    float mx = fmaxf(z0, fmaxf(z1, z2));
    float part = mx + __logf(__expf(z0 - mx) + __expf(z1 - mx) + __expf(z2 - mx));
    float tot = block_reduce256(active ? part : 0.f);
    if (tid == 0) partial[partialOffset + blockIdx.x] = tot;
}

// ---------------------------------------------------------------------------
// Kernel C: combine partials -> out[0] = (sum)/B
// ---------------------------------------------------------------------------
__global__ void crf_finalize_kernel(const float* __restrict__ partial, int n,
                                    float invB, float* __restrict__ out)
{
    int tid = threadIdx.x;
    float a = 0.f;
    for (int i = tid; i < n; i += 256) a += partial[i];
    float tot = block_reduce256(a);
    if (tid == 0) out[0] = tot * invB;
}

extern "C" void kernel_launch(void* const* d_in, const int* in_sizes, int n_in,
                              void* d_out, int out_size, void* d_ws, size_t ws_size,
                              hipStream_t stream) {
    const float* em     = (const float*)d_in[0];
    const int*   tg     = (const int*)  d_in[1];
    const int*   mk     = (const int*)  d_in[2];
    const int*   pid    = (const int*)  d_in[3];
    const int*   the    = (const int*)  d_in[4];
    const int*   intens = (const int*)  d_in[5];
    const int*   lower  = (const int*)  d_in[6];
    const float* pw     = (const float*)d_in[7];
    const float* T      = (const float*)d_in[8];
    const float* startT = (const float*)d_in[9];
    const float* endT   = (const float*)d_in[10];

    const int B = in_sizes[1] / SEQ_LEN;   // tags is (B, S)
    float* ws = (float*)d_ws;

    const int GRID_A = 2048;
    crf_score_constraint_kernel<<<GRID_A, 256, 0, stream>>>(
        em, tg, mk, pid, the, intens, lower, pw, T, startT, endT, ws, B);

    const int GRID_B = (B + 255) / 256;
    crf_partition_kernel<<<GRID_B, 256, 0, stream>>>(
        em, mk, pid, pw, T, startT, endT, ws, B, GRID_A);

    crf_finalize_kernel<<<1, 256, 0, stream>>>(
        ws, GRID_A + GRID_B, 1.0f / (float)B, (float*)d_out);
}